// GroupNormalNorm_5377299054677
// MI455X (gfx1250) — compile-verified
//
#include <hip/hip_runtime.h>
#include <cstdint>

// GroupNormalNorm for x:(32,128,128,128) f32, G=32.
// 1024 groups x 65536 floats. One workgroup (1024 thr = 32 waves) per group.
// Group tile (256 KB) staged in the 320 KB WGP LDS via CDNA5 async
// global->LDS b128 loads (NT-tagged), then 3 passes (mean/var, newton sums,
// power transform) run out of LDS with b128 accesses. Output written with
// b128 nontemporal stores. Global traffic = 1 read + 1 write = roofline.

#define GROUPS_TOTAL 1024
#define GROUP_N      65536
#define THREADS      1024
#define NWAVES       32
#define EPSF         1e-5f
#define EPS_SQRTF    0.0031622776601683794f
#define LN2F         0.6931471805599453f

typedef __attribute__((ext_vector_type(4))) float v4f;

__device__ __forceinline__ float wave_sum(float v) {
#pragma unroll
  for (int off = 16; off > 0; off >>= 1)
    v += __shfl_xor(v, off, 32);
  return v;
}

// Sum K per-thread values across the 1024-thread workgroup; result returned
// to every thread in vals[].
template <int K>
__device__ __forceinline__ void block_sum(float* vals, float* scratch, int tid) {
  const int wave = tid >> 5;
  const int lane = tid & 31;
  __syncthreads();  // protect scratch from any previous use
#pragma unroll
  for (int k = 0; k < K; ++k) vals[k] = wave_sum(vals[k]);
  if (lane == 0) {
#pragma unroll
    for (int k = 0; k < K; ++k) scratch[k * NWAVES + wave] = vals[k];
  }
  __syncthreads();
  if (wave == 0) {
#pragma unroll
    for (int k = 0; k < K; ++k) {
      float t = scratch[k * NWAVES + lane];
      t = wave_sum(t);
      if (lane == 0) scratch[k * NWAVES] = t;
    }
  }
  __syncthreads();
#pragma unroll
  for (int k = 0; k < K; ++k) vals[k] = scratch[k * NWAVES];
}

// One element of the Newton-sum pass: accumulate the 5 per-group sums.
__device__ __forceinline__ void newton_accum(float v, float mean, float inv_std,
                                             float* s1, float* sd, float* sxd,
                                             float* sxd2, float* sdd) {
  float xn  = (v - mean) * inv_std;
  float s   = (xn >= 0.f) ? 1.f : -1.f;          // _signed: sign(0) -> 1
  float a   = fabsf(xn);
  float ap1 = 1.f + a;
  float l   = LN2F * __builtin_amdgcn_logf(ap1); // log1p via v_log_f32
  float d   = fmaf(ap1, l, -a);                  // (1+a)*l - a
  *s1  += s * l;
  *sd  += d;
  *sxd  = fmaf(xn, d, *sxd);
  float p1 = fmaf(ap1 * l, l, -2.f * d);         // (1+a)*l^2 - 2d
  *sxd2 = fmaf(xn * s, p1, *sxd2);               // xn * d2, d2 = s*p1
  *sdd  = fmaf(d, d, *sdd);
}

// One element of the power-transform pass.
__device__ __forceinline__ float transform1(float v, float mean, float inv_std,
                                            float lm1, float w, float b) {
  float xn  = (v - mean) * inv_std;
  float s   = (xn >= 0.f) ? 1.f : -1.f;
  float a   = fabsf(xn);
  float L2  = __builtin_amdgcn_logf(1.f + a);      // log2(1+a)
  float eta = fmaf(s, lm1, 1.f);
  float es  = (eta >= 0.f) ? EPS_SQRTF : -EPS_SQRTF;
  float ex  = eta + es;                            // eta + sign(eta)*sqrt(eps)
  float p   = __builtin_amdgcn_exp2f(ex * L2);     // (1+a)^ex
  float tr1 = s * __builtin_amdgcn_rcpf(ex) * (p - 1.f);
  float tr2 = s * (LN2F * L2);
  float tr  = (fabsf(eta) <= EPS_SQRTF) ? tr2 : tr1;
  return fmaf(tr, w, b);
}

__global__ __launch_bounds__(THREADS, 1)
void gnn_kernel(const float* __restrict__ x, const float* __restrict__ weight,
                const float* __restrict__ bias, float* __restrict__ out) {
  __shared__ float tile[GROUP_N];        // 256 KB of the 320 KB WGP LDS
  __shared__ float scratch[5 * NWAVES];  // reduction partials

  const int tid  = threadIdx.x;
  const int grp  = blockIdx.x;   // b*32 + g
  const int g    = grp & 31;     // group within batch -> channels [4g, 4g+4)
  const size_t base = (size_t)grp * GROUP_N;

  // ---------- Phase A: async global -> LDS copy (16 B / lane / round) ----------
  {
    const float* gsrc = x + base;
#pragma unroll
    for (int j = 0; j < 16; ++j) {
      const int i4 = j * THREADS + tid;                       // float4 index
      uint32_t lds_addr = (uint32_t)(uintptr_t)(&tile[i4 * 4]);
      const float* gp = gsrc + (size_t)i4 * 4;
      asm volatile("global_load_async_to_lds_b128 %0, %1, off th:TH_LOAD_NT"
                   :: "v"(lds_addr), "v"(gp)
                   : "memory");
    }
    asm volatile("s_wait_asynccnt 0x0" ::: "memory");
  }
  __syncthreads();

  const v4f* tile4 = reinterpret_cast<const v4f*>(tile);

  // ---------- Phase B: mean / var (b128 LDS reads) ----------
  float acc[5];
  acc[0] = 0.0f; acc[1] = 0.0f;
#pragma unroll 4
  for (int j = 0; j < 16; ++j) {
    v4f v = tile4[j * THREADS + tid];    // lane t -> word 4t: conflict-free b128
#pragma unroll
    for (int e = 0; e < 4; ++e) {
      float f = v[e];
      acc[0] += f;
      acc[1] = fmaf(f, f, acc[1]);
    }
  }
  block_sum<2>(acc, scratch, tid);
  const float invN    = 1.0f / (float)GROUP_N;
  const float mean    = acc[0] * invN;
  const float var     = fmaf(-mean, mean, acc[1] * invN);
  const float inv_std = __builtin_amdgcn_rsqf(var + EPSF);   // v_rsq_f32

  // ---------- Phase C: Newton sums for lambda ----------
  float s1 = 0.f, sd = 0.f, sxd = 0.f, sxd2 = 0.f, sdd = 0.f;
#pragma unroll 2
  for (int j = 0; j < 16; ++j) {
    v4f v = tile4[j * THREADS + tid];
#pragma unroll
    for (int e = 0; e < 4; ++e)
      newton_accum(v[e], mean, inv_std, &s1, &sd, &sxd, &sxd2, &sdd);
  }
  acc[0] = s1; acc[1] = sd; acc[2] = sxd; acc[3] = sxd2; acc[4] = sdd;
  block_sum<5>(acc, scratch, tid);

  const float s1m   = acc[0] * invN;
  const float dmean = acc[1] * invN;
  const float dvar  = 2.f * acc[2] * invN;
  const float g1    = 0.5f * dvar - s1m;
  const float d2var = 2.f * (acc[3] * invN + fmaf(-dmean, dmean, acc[4] * invN));
  const float g2    = fmaf(-0.5f * dvar, dvar, 0.5f * d2var);
  const float lmbda = 1.f - g1 * __builtin_amdgcn_rcpf(g2 + EPS_SQRTF);
  const float lm1   = lmbda - 1.f;

  // ---------- Phase D: power transform + per-channel affine ----------
  // float4 index i4 = j*1024 + tid covers floats [j*4096, (j+1)*4096):
  // each j-iteration lies in one 16384-float channel block -> channel = j>>2.
  float wv[4], bv[4];
#pragma unroll
  for (int c = 0; c < 4; ++c) {
    wv[c] = weight[g * 4 + c];
    bv[c] = bias[g * 4 + c];
  }
  v4f* out4 = reinterpret_cast<v4f*>(out + base);
#pragma unroll
  for (int c = 0; c < 4; ++c) {          // channel uniform per j-iteration
#pragma unroll
    for (int j2 = 0; j2 < 4; ++j2) {
      const int i4 = (c * 4 + j2) * THREADS + tid;
      v4f v = tile4[i4];
      v4f o;
#pragma unroll
      for (int e = 0; e < 4; ++e)
        o[e] = transform1(v[e], mean, inv_std, lm1, wv[c], bv[c]);
      __builtin_nontemporal_store(o, out4 + i4);   // b128 NT store
    }
  }
}

extern "C" void kernel_launch(void* const* d_in, const int* in_sizes, int n_in,
                              void* d_out, int out_size, void* d_ws, size_t ws_size,
                              hipStream_t stream) {
  (void)in_sizes; (void)n_in; (void)d_ws; (void)ws_size; (void)out_size;
  const float* x = (const float*)d_in[0];
  const float* w = (const float*)d_in[1];
  const float* b = (const float*)d_in[2];
  float* out = (float*)d_out;
  gnn_kernel<<<dim3(GROUPS_TOTAL), dim3(THREADS), 0, stream>>>(x, w, b, out);
}